// Attention_19086834663633
// MI455X (gfx1250) — compile-verified
//
#include <hip/hip_runtime.h>

typedef float v2f __attribute__((ext_vector_type(2)));
typedef float v4f __attribute__((ext_vector_type(4)));
typedef float v8f __attribute__((ext_vector_type(8)));

#define SEQ    4096
#define BATCH  64
#define HIDDEN 1024
#define S_TILE 128   // s-rows per block (8 waves x 16)
#define H_C    64    // h chunk staged in LDS
#define H_P    68    // padded LDS row stride in dwords (16B aligned, conflict-free)
#define LD_PER_THREAD 8  // (S_TILE * H_C/4) / 256 float4 loads per thread

// ---------------------------------------------------------------------------
// Kernel 1: scores[b, s] = dot(q[b,:], K[s,b,:]) via V_WMMA_F32_16X16X4_F32.
// A = 16 key rows x 4 h (documented layout), B = q chunk broadcast over all
// 16 N columns -> every column of D holds the same per-row score.
// Keys are streamed once -> non-temporal loads; staged through LDS with 8
// loads in flight per thread for memory-level parallelism.
// ---------------------------------------------------------------------------
__global__ __launch_bounds__(256) void scores_kernel(const float* __restrict__ q,
                                                     const float* __restrict__ k,
                                                     float* __restrict__ out) {
    __shared__ float kbuf[S_TILE * H_P];
    __shared__ float qbuf[H_P];

    const int b      = blockIdx.y;
    const int s_base = blockIdx.x * S_TILE;
    const int tid    = threadIdx.x;
    const int lane   = tid & 31;
    const int wave   = tid >> 5;
    const int m      = lane & 15;          // A-matrix row within wave tile
    const int c2     = (lane >> 4) << 1;   // K offset: 0 (lanes 0-15) / 2 (lanes 16-31)

    v8f acc = {};

    const float* kb = k + (size_t)b * HIDDEN;   // batch offset into keys
    const float* qb = q + (size_t)b * HIDDEN;

    for (int h0 = 0; h0 < HIDDEN; h0 += H_C) {
        __syncthreads();   // previous chunk's compute done before LDS overwrite

        // ---- stage: issue all global loads first (8 in flight / thread) ----
        v4f kv[LD_PER_THREAD];
        #pragma unroll
        for (int j = 0; j < LD_PER_THREAD; ++j) {
            const int i    = tid + j * 256;         // 0 .. 2047
            const int row  = i >> 4;                // 16 float4 per row
            const int col4 = i & 15;
            kv[j] = __builtin_nontemporal_load((const v4f*)(kb +
                (size_t)(s_base + row) * (BATCH * HIDDEN) + h0 + col4 * 4));
        }
        v4f qv;
        if (tid < H_C / 4)
            qv = *(const v4f*)(qb + h0 + tid * 4);   // q: keep cached (reused)

        // ---- drain into LDS ----
        #pragma unroll
        for (int j = 0; j < LD_PER_THREAD; ++j) {
            const int i    = tid + j * 256;
            const int row  = i >> 4;
            const int col4 = i & 15;
            *(v4f*)(kbuf + row * H_P + col4 * 4) = kv[j];
        }
        if (tid < H_C / 4)
            *(v4f*)(qbuf + tid * 4) = qv;
        __syncthreads();

        // ---- compute: 16 chained f32 WMMAs over the 64-wide h chunk ----
        const float* arow = kbuf + (wave * 16 + m) * H_P + c2;
        const float* brow = qbuf + c2;
        #pragma unroll
        for (int kk = 0; kk < H_C / 4; ++kk) {
            v2f a  = *(const v2f*)(arow + kk * 4);   // K[s_row][h0+4kk+c2 .. +1]
            v2f bq = *(const v2f*)(brow + kk * 4);   // q[h0+4kk+c2 .. +1] (bcast N)
            acc = __builtin_amdgcn_wmma_f32_16x16x4_f32(
                /*neg_a=*/false, a, /*neg_b=*/false, bq,
                /*c_mod=*/(short)0, acc, /*reuse_a=*/false, /*reuse_b=*/false);
        }
    }

    // D layout: VGPR g, lanes 0-15 -> row M=g (N=lane), lanes 16-31 -> row M=g+8.
    // All N columns identical; lane 0 writes rows g, lane 16 writes rows g+8.
    if ((lane & 15) == 0) {
        const int half = lane >> 4;   // 0 or 1
        float* o = out + (size_t)b * SEQ + s_base + wave * 16 + half * 8;
        #pragma unroll
        for (int g = 0; g < 8; ++g) o[g] = acc[g];
    }
}

// ---------------------------------------------------------------------------
// Kernel 2: in-place softmax over S=4096 per batch row (one block per b).
// Scores (1 MB) are L2-resident; cost is negligible vs the 1 GiB key stream.
// ---------------------------------------------------------------------------
__global__ __launch_bounds__(256) void softmax_kernel(float* __restrict__ out) {
    __shared__ float red[256];
    const int b   = blockIdx.x;
    const int tid = threadIdx.x;
    float* row = out + (size_t)b * SEQ;

    float v[16];
    float mx = -INFINITY;
    #pragma unroll
    for (int i = 0; i < 16; ++i) {          // coalesced strided reads
        v[i] = row[tid + i * 256];
        mx = fmaxf(mx, v[i]);
    }
    red[tid] = mx;
    __syncthreads();
    for (int s = 128; s > 0; s >>= 1) {
        if (tid < s) red[tid] = fmaxf(red[tid], red[tid + s]);
        __syncthreads();
    }
    mx = red[0];
    __syncthreads();

    float sum = 0.f;
    #pragma unroll
    for (int i = 0; i < 16; ++i) { v[i] = __expf(v[i] - mx); sum += v[i]; }
    red[tid] = sum;
    __syncthreads();
    for (int s = 128; s > 0; s >>= 1) {
        if (tid < s) red[tid] += red[tid + s];
        __syncthreads();
    }
    const float inv = 1.f / red[0];
    #pragma unroll
    for (int i = 0; i < 16; ++i) row[tid + i * 256] = v[i] * inv;
}

extern "C" void kernel_launch(void* const* d_in, const int* in_sizes, int n_in,
                              void* d_out, int out_size, void* d_ws, size_t ws_size,
                              hipStream_t stream) {
    (void)in_sizes; (void)n_in; (void)out_size; (void)d_ws; (void)ws_size;
    const float* q = (const float*)d_in[0];   // [1, B, H]
    const float* k = (const float*)d_in[1];   // [S, B, H]
    float* out = (float*)d_out;               // [1, B, S] == B*S floats

    dim3 grid1(SEQ / S_TILE, BATCH);          // (32, 64)
    scores_kernel<<<grid1, 256, 0, stream>>>(q, k, out);
    softmax_kernel<<<BATCH, 256, 0, stream>>>(out);
}